// SparseTransition_16673063043300
// MI455X (gfx1250) — compile-verified
//
#include <hip/hip_runtime.h>
#include <math.h>

// ---------------------------------------------------------------------------
// SparseTransition fused kernel for MI455X (gfx1250, wave32)
//
// Shapes: B=4, J=4096, K=8192, D=64, Dd=64, TOPK=64
//   logits  = val @ W_route  (fp32 WMMA 16x16x4, 16 chained per 16-col tile)
//   routes  = exact top-64 masked softmax (histogram-pruned radix select)
//   pval    = val @ W_val    (fp32 WMMA)
//   out_state = dst_state + sum_j routes*softplus(state)*state
//   out_val   = dst_val   + sum_j routes*softplus(state)*pval
//
// One workgroup = one (b, 16-row j tile); 256 threads = 8 waves.
// Logits kept in LDS in two 4096-col halves (256KB of the 320KB WGP LDS).
// A tile staged via global_load_async_to_lds_b128 (ASYNCcnt path).
// ---------------------------------------------------------------------------

typedef __attribute__((ext_vector_type(2))) float v2f;
typedef __attribute__((ext_vector_type(8))) float v8f;

#define B_   4
#define J_   4096
#define K_   8192
#define D_   64
#define DD_  64
#define TOPK_ 64
#define ROWS 16
#define HALF 4096
#define NTHREADS 256
#define NWAVES 8
#define CAP_  160   // candidate scratch capacity per row

// ---- LDS layout (bytes) ----------------------------------------------------
#define OFF_LOGIT 0
#define OFF_PVAL  (OFF_LOGIT + ROWS*HALF*4)       // 262144
#define OFF_ATILE (OFF_PVAL  + ROWS*DD_*4)        // +4096
#define OFF_HIST  (OFF_ATILE + ROWS*D_*4)         // +4096
#define OFF_SCRV  (OFF_HIST  + NWAVES*256*4)      // +8192
#define OFF_SCRI  (OFF_SCRV  + ROWS*CAP_*4)       // +10240
#define OFF_CANDV (OFF_SCRI  + ROWS*CAP_*4)       // +10240
#define OFF_CANDI (OFF_CANDV + ROWS*128*4)        // +8192
#define OFF_FINV  (OFF_CANDI + ROWS*128*4)        // +8192
#define OFF_FINI  (OFF_FINV  + ROWS*TOPK_*4)      // +4096
#define OFF_CNT   (OFF_FINI  + ROWS*TOPK_*4)      // +4096
#define LDS_BYTES (OFF_CNT   + NWAVES*4)          // 323616 < 327680

// Monotonic u32 key: a > b (float) <=> ku(a) > ku(b)
__device__ __forceinline__ unsigned ku(float f) {
  unsigned u = __float_as_uint(f);
  return u ^ ((unsigned)((int)u >> 31) | 0x80000000u);
}

// Wave-local LDS fence: same-wave DS ops retire in order (DScnt); stop the
// compiler/issue from hoisting LDS reads above pending ds_add traffic.
__device__ __forceinline__ void wfence() {
  __builtin_amdgcn_wave_barrier();
  asm volatile("s_wait_dscnt 0" ::: "memory");
  __builtin_amdgcn_wave_barrier();
}

__device__ __forceinline__ float softplusf(float x) {
  return x > 20.f ? x : log1pf(__expf(x));
}

// Exact rank-`want` threshold key over a[0..n) (LDS), per-wave, 4x8-bit radix.
__device__ unsigned radix_select(const float* a, int n, unsigned* hist,
                                 int lane, int want, int* rankEq) {
  unsigned prefix = 0;
  int rank = want;
  for (int level = 0; level < 4; ++level) {
    const int shift = 24 - 8 * level;
    for (int i = lane; i < 256; i += 32) hist[i] = 0u;
    wfence();
    for (int i = lane; i < n; i += 32) {
      unsigned u = ku(a[i]);
      bool match = (level == 0) || ((u >> (shift + 8)) == prefix);
      if (match) atomicAdd(&hist[(u >> shift) & 255u], 1u);
    }
    wfence();
    int acc = 0; unsigned chosen = 0;
    for (int bin = 255; bin >= 0; --bin) {     // redundant broadcast scan
      int c = (int)hist[bin];
      if (acc + c >= rank) { chosen = (unsigned)bin; break; }
      acc += c;
    }
    rank -= acc;
    prefix = (prefix << 8) | chosen;
  }
  *rankEq = rank;
  return prefix;
}

// Compact entries with key>T, then key==T (ties, arbitrary) up to TOPK_ total.
__device__ void collect_topk(const float* a, int n, unsigned T, unsigned* cnt,
                             int lane, float* dstV, int* dstI,
                             const int* srcI, int idxBase) {
  if (lane == 0) *cnt = 0u;
  wfence();
  for (int i = lane; i < n; i += 32) {
    float v = a[i];
    if (ku(v) > T) {
      int s = (int)atomicAdd(cnt, 1u);
      dstV[s] = v;
      dstI[s] = srcI ? srcI[i] : (idxBase + i);
    }
  }
  wfence();
  for (int i = lane; i < n; i += 32) {
    float v = a[i];
    if (ku(v) == T) {
      int s = (int)atomicAdd(cnt, 1u);
      if (s < TOPK_) {
        dstV[s] = v;
        dstI[s] = srcI ? srcI[i] : (idxBase + i);
      }
    }
  }
  wfence();
}

// Exact top-64 of a[0..n): ONE coarse histogram pass isolates a small
// candidate set (logits ~N(0,1): the >=rank-64 tail lives in 1-2 exponent
// bins, ~100 values), then an exact radix select runs on <=CAP_ candidates.
// Falls back to deeper levels only if a bin is pathologically dense; full-key
// ties at level 3 are exact ties (arbitrary pick matches lax.top_k).
__device__ void select_row_topk(const float* a, int n, int kb,
                                unsigned* hist, unsigned* cnt, int lane,
                                float* scrV, int* scrI,
                                float* outV, int* outI) {
  unsigned prefix = 0, Tlow = 0;
  int rank = TOPK_, accTot = 0;
  for (int level = 0; level < 4; ++level) {
    const int shift = 24 - 8 * level;
    for (int i = lane; i < 256; i += 32) hist[i] = 0u;
    wfence();
    for (int i = lane; i < n; i += 32) {
      unsigned u = ku(a[i]);
      bool match = (level == 0) || ((u >> (shift + 8)) == prefix);
      if (match) atomicAdd(&hist[(u >> shift) & 255u], 1u);
    }
    wfence();
    int acc = 0; unsigned chosen = 0; int cc = 0;
    for (int bin = 255; bin >= 0; --bin) {
      int c = (int)hist[bin];
      if (acc + c >= rank) { chosen = (unsigned)bin; cc = c; break; }
      acc += c;
    }
    rank -= acc;
    accTot += acc;
    prefix = (prefix << 8) | chosen;
    Tlow = prefix << shift;                     // floor of chosen bin
    if (accTot + cc <= CAP_ || level == 3) break;
  }
  // collect candidate set: key >= Tlow (size in [64, CAP_] except tie floods)
  if (lane == 0) *cnt = 0u;
  wfence();
  for (int i = lane; i < n; i += 32) {
    float v = a[i];
    if (ku(v) >= Tlow) {
      int s = (int)atomicAdd(cnt, 1u);
      if (s < CAP_) { scrV[s] = v; scrI[s] = kb + i; }
    }
  }
  wfence();
  int nc = (int)*cnt; if (nc > CAP_) nc = CAP_;
  // exact top-64 among nc candidates
  int rankEq;
  unsigned T = radix_select(scrV, nc, hist, lane, TOPK_, &rankEq);
  collect_topk(scrV, nc, T, cnt, lane, outV, outI, scrI, 0);
}

__global__ __launch_bounds__(NTHREADS, 1)
void st_fused_kernel(const float* __restrict__ val,
                     const float* __restrict__ state,
                     const float* __restrict__ W_route,
                     const float* __restrict__ W_val,
                     float* __restrict__ outState,
                     float* __restrict__ outVal) {
  extern __shared__ char smem[];
  float*    lgt     = (float*)(smem + OFF_LOGIT);   // [16][4096]
  float*    pval    = (float*)(smem + OFF_PVAL);    // [16][64]
  float*    aT      = (float*)(smem + OFF_ATILE);   // [16][64]
  unsigned* histAll = (unsigned*)(smem + OFF_HIST); // [8][256]
  float*    scrV    = (float*)(smem + OFF_SCRV);    // [16][CAP_]
  int*      scrI    = (int*)(smem + OFF_SCRI);
  float*    candV   = (float*)(smem + OFF_CANDV);   // [16][128]
  int*      candI   = (int*)(smem + OFF_CANDI);
  float*    finV    = (float*)(smem + OFF_FINV);    // [16][64]
  int*      finI    = (int*)(smem + OFF_FINI);
  unsigned* cntAll  = (unsigned*)(smem + OFF_CNT);  // [8]

  const int tid  = threadIdx.x;
  const int wave = tid >> 5;
  const int lane = tid & 31;
  const int b    = blockIdx.x / (J_ / ROWS);
  const int j0   = (blockIdx.x % (J_ / ROWS)) * ROWS;

  unsigned* hist = histAll + wave * 256;
  unsigned* cnt  = cntAll + wave;

  // ---- Stage 1: async-stage A tile (16 rows x 64 dims, 4KB) into LDS ------
  // 256 threads x 16B = exactly 4096B; ASYNCcnt-tracked DMA into LDS.
  {
    unsigned ldsOff = (unsigned)(OFF_ATILE) + (unsigned)tid * 16u;
    const float* g = val + ((size_t)b * J_ + j0) * D_ + (size_t)tid * 4;
    asm volatile("global_load_async_to_lds_b128 %0, %1, off"
                 :: "v"(ldsOff), "v"(g) : "memory");
    asm volatile("s_wait_asynccnt 0" ::: "memory");
  }
  __syncthreads();

  // A fragments for V_WMMA_F32_16X16X4_F32 (ISA 7.12.2, 32-bit A 16x4):
  // lanes 0-15: M=lane, VGPR0=K0,VGPR1=K1 ; lanes 16-31: M=lane-16, K2/K3.
  const int m     = lane & 15;
  const int khalf = (lane >> 4) << 1;   // 0 or 2
  v2f aF[16];
#pragma unroll
  for (int s = 0; s < 16; ++s) {
    aF[s].x = aT[m * D_ + 4 * s + khalf];
    aF[s].y = aT[m * D_ + 4 * s + khalf + 1];
  }

  // ---- Stage 2: pval = valTile @ W_val (waves 0..3, one 16-col tile each) --
  if (wave < 4) {
    const int n0 = wave * 16;
    const float* bp = W_val + (size_t)khalf * DD_ + n0 + m;
    v8f c = {};
#pragma unroll
    for (int s = 0; s < 16; ++s) {
      v2f bF;
      bF.x = bp[(size_t)(4 * s) * DD_];
      bF.y = bp[(size_t)(4 * s + 1) * DD_];
      c = __builtin_amdgcn_wmma_f32_16x16x4_f32(false, aF[s], false, bF,
                                                (short)0, c, false, false);
    }
#pragma unroll
    for (int r = 0; r < 8; ++r) {
      int row = r + ((lane >> 4) << 3);   // C/D layout: M = r (+8 for hi lanes)
      pval[row * DD_ + n0 + m] = c[r];
    }
  }
  __syncthreads();

  const float stv0 = state[(size_t)b * J_ + j0 + wave];
  const float stv1 = state[(size_t)b * J_ + j0 + wave + 8];

  // ---- Stage 3: two K halves: WMMA logits -> LDS -> exact top-64 ----------
  for (int h = 0; h < 2; ++h) {
    const int kb = h * HALF;

    // 256 column tiles per half, 32 per wave, 16 chained WMMA per tile
    for (int t = 0; t < HALF / 16 / NWAVES; ++t) {
      const int tt  = t * NWAVES + wave;
      const int n0l = tt * 16;
      const float* bp = W_route + (size_t)khalf * K_ + (kb + n0l) + m;
      __builtin_prefetch(bp + 16 * NWAVES, 0, 1);   // next tile's B panel
      v8f c = {};
#pragma unroll
      for (int s = 0; s < 16; ++s) {
        v2f bF;
        bF.x = bp[(size_t)(4 * s) * K_];
        bF.y = bp[(size_t)(4 * s + 1) * K_];
        c = __builtin_amdgcn_wmma_f32_16x16x4_f32(false, aF[s], false, bF,
                                                  (short)0, c, false, false);
      }
#pragma unroll
      for (int r = 0; r < 8; ++r) {
        int row = r + ((lane >> 4) << 3);
        lgt[row * HALF + n0l + m] = c[r];
      }
    }
    __syncthreads();

    // per-wave exact top-64 of this half for rows {wave, wave+8}
    for (int rr = 0; rr < 2; ++rr) {
      const int r = wave + rr * 8;
      select_row_topk(lgt + (size_t)r * HALF, HALF, kb, hist, cnt, lane,
                      scrV + r * CAP_, scrI + r * CAP_,
                      candV + r * 128 + h * TOPK_,
                      candI + r * 128 + h * TOPK_);
    }
    __syncthreads();   // logits LDS reused by next half
  }

  // ---- Stage 4: merge 128 -> 64, softmax, scatter-add ---------------------
  for (int rr = 0; rr < 2; ++rr) {
    const int r = wave + rr * 8;
    int rankEq;
    unsigned T = radix_select(candV + r * 128, 128, hist, lane, TOPK_, &rankEq);
    collect_topk(candV + r * 128, 128, T, cnt, lane,
                 finV + r * TOPK_, finI + r * TOPK_, candI + r * 128, 0);

    // masked softmax over exactly the 64 kept entries (2 per lane)
    float v0 = finV[r * TOPK_ + lane];
    float v1 = finV[r * TOPK_ + lane + 32];
    float mx = fmaxf(v0, v1);
#pragma unroll
    for (int ofs = 16; ofs > 0; ofs >>= 1) mx = fmaxf(mx, __shfl_xor(mx, ofs, 32));
    float e0 = __expf(v0 - mx), e1 = __expf(v1 - mx);
    float sum = e0 + e1;
#pragma unroll
    for (int ofs = 16; ofs > 0; ofs >>= 1) sum += __shfl_xor(sum, ofs, 32);
    finV[r * TOPK_ + lane]      = e0;
    finV[r * TOPK_ + lane + 32] = e1;
    wfence();

    const float stv  = rr ? stv1 : stv0;
    const float coef = softplusf(stv) / sum;    // softplus(state) / softmax Z
    const float p0 = pval[r * DD_ + lane];
    const float p1 = pval[r * DD_ + lane + 32];
    for (int cI = 0; cI < TOPK_; ++cI) {
      float w = finV[r * TOPK_ + cI] * coef;    // weighted[b,j,k]
      int   k = finI[r * TOPK_ + cI];
      float* dv = outVal + ((size_t)b * K_ + k) * DD_;
      atomicAdd(dv + lane,      w * p0);
      atomicAdd(dv + lane + 32, w * p1);
      if (lane == 0) atomicAdd(outState + (size_t)b * K_ + k, w * stv);
    }
  }
}

// Initialize d_out with dst_state / dst_val (merge_mode='add' baseline).
__global__ void st_init_out(const float* __restrict__ ds,
                            const float* __restrict__ dv,
                            float* __restrict__ outState,
                            float* __restrict__ outVal) {
  const int stride = gridDim.x * blockDim.x;
  const int i0 = blockIdx.x * blockDim.x + threadIdx.x;
  for (int i = i0; i < B_ * K_; i += stride) outState[i] = ds[i];
  for (int i = i0; i < B_ * K_ * DD_; i += stride) outVal[i] = dv[i];
}

extern "C" void kernel_launch(void* const* d_in, const int* in_sizes, int n_in,
                              void* d_out, int out_size, void* d_ws, size_t ws_size,
                              hipStream_t stream) {
  const float* val       = (const float*)d_in[0];
  const float* state     = (const float*)d_in[1];
  const float* dst_state = (const float*)d_in[2];
  const float* dst_val   = (const float*)d_in[3];
  const float* W_route   = (const float*)d_in[4];
  const float* W_val     = (const float*)d_in[5];
  float* outState = (float*)d_out;                 // [B,K]
  float* outVal   = outState + B_ * K_;            // [B,K,Dd]

  (void)hipFuncSetAttribute((const void*)st_fused_kernel,
                            hipFuncAttributeMaxDynamicSharedMemorySize,
                            LDS_BYTES);

  st_init_out<<<512, 256, 0, stream>>>(dst_state, dst_val, outState, outVal);
  st_fused_kernel<<<B_ * (J_ / ROWS), NTHREADS, LDS_BYTES, stream>>>(
      val, state, W_route, W_val, outState, outVal);
}